// GraphSAGE_38293928411634
// MI455X (gfx1250) — compile-verified
//
#include <hip/hip_runtime.h>

#define N_NODES   100000
#define N_EDGES   2500000
#define IN_CH     32
#define HID       16
#define OUT_CH    64
#define NUM_GRAPHS 64
#define N_TILES   (N_NODES / 16)   // 6250, exact

typedef __attribute__((ext_vector_type(2))) float v2f;
typedef __attribute__((ext_vector_type(8))) float v8f;

// workspace layout in floats
#define WS_DEG   0                         // N_NODES
#define WS_AGG1  102400                    // N_NODES*32
#define WS_H     (WS_AGG1 + 3200000)       // N_NODES*16
#define WS_AGG2  (WS_H + 1600000)          // N_NODES*16
#define WS_GSUM  (WS_AGG2 + 1600000)       // 64*64
#define WS_GCNT  (WS_GSUM + 4096)          // 64
#define WS_TOTAL (WS_GCNT + 64)

// ---------------------------------------------------------------------------
// Edge scatter for layer 1: agg1[dst] += x[src], deg[dst] += 1
// one thread per (edge, channel); feature tables are L2-resident on MI455X.
// ---------------------------------------------------------------------------
__global__ void sage_scatter1(const float* __restrict__ x,
                              const long long* __restrict__ src,
                              const long long* __restrict__ dst,
                              float* __restrict__ agg1,
                              float* __restrict__ deg) {
    long long t = (long long)blockIdx.x * blockDim.x + threadIdx.x;
    if (t >= (long long)N_EDGES * IN_CH) return;
    int e = (int)(t >> 5);
    int c = (int)(t & 31);
    int s = (int)src[e];
    int d = (int)dst[e];
    atomicAdd(&agg1[(long long)d * IN_CH + c], x[(long long)s * IN_CH + c]);
    if (c == 0) atomicAdd(&deg[d], 1.0f);
}

// ---------------------------------------------------------------------------
// Layer 1: h = relu( mean(agg1) @ w1_l^T + b1 + x @ w1_r^T )
// One wave32 per 16-node tile. fp32 WMMA 16x16x4, K=32 -> 8 steps per matrix.
// A fragment (16x4 f32): lanes 0-15 hold K=k,k+1 ; lanes 16-31 hold K=k+2,k+3
// (row m = lane&15). B fragment symmetric with N = lane&15.
// C (16x16 f32): lane n = lane&15, VGPR j = row j + 8*(lane>=16).
// ---------------------------------------------------------------------------
__global__ void sage_layer1(const float* __restrict__ x,
                            const float* __restrict__ agg1,
                            const float* __restrict__ deg,
                            const float* __restrict__ w1l,   // [16][32]
                            const float* __restrict__ b1,    // [16]
                            const float* __restrict__ w1r,   // [16][32]
                            float* __restrict__ h) {
    int wave = (blockIdx.x * blockDim.x + threadIdx.x) >> 5;
    int lane = threadIdx.x & 31;
    if (wave >= N_TILES) return;
    int m     = lane & 15;
    int khalf = lane >> 4;            // 0: K=k,k+1   1: K=k+2,k+3
    int node_m = wave * 16 + m;
    float rdeg = 1.0f / fmaxf(deg[node_m], 1.0f);

    const float* aggrow = agg1 + (long long)node_m * IN_CH;
    const float* xrow   = x    + (long long)node_m * IN_CH;

    v8f c = {};
#pragma unroll
    for (int k = 0; k < IN_CH; k += 4) {
        int ko = k + 2 * khalf;
        v2f a = *(const v2f*)(aggrow + ko);
        a.x *= rdeg; a.y *= rdeg;                       // fused mean
        v2f b = *(const v2f*)(w1l + m * IN_CH + ko);    // B[k][n] = w[n][k]
        c = __builtin_amdgcn_wmma_f32_16x16x4_f32(false, a, false, b,
                                                  (short)0, c, false, false);
    }
#pragma unroll
    for (int k = 0; k < IN_CH; k += 4) {
        int ko = k + 2 * khalf;
        v2f a = *(const v2f*)(xrow + ko);
        v2f b = *(const v2f*)(w1r + m * IN_CH + ko);
        c = __builtin_amdgcn_wmma_f32_16x16x4_f32(false, a, false, b,
                                                  (short)0, c, false, false);
    }

    float bias = b1[m];               // this lane's column n == lane&15
#pragma unroll
    for (int j = 0; j < 8; ++j) {
        int row  = j + 8 * khalf;
        int node = wave * 16 + row;
        h[(long long)node * HID + m] = fmaxf(c[j] + bias, 0.0f);
    }
}

// ---------------------------------------------------------------------------
// Edge scatter for layer 2: agg2[dst] += h[src]   (16 channels)
// ---------------------------------------------------------------------------
__global__ void sage_scatter2(const float* __restrict__ h,
                              const long long* __restrict__ src,
                              const long long* __restrict__ dst,
                              float* __restrict__ agg2) {
    long long t = (long long)blockIdx.x * blockDim.x + threadIdx.x;
    if (t >= (long long)N_EDGES * HID) return;
    int e = (int)(t >> 4);
    int c = (int)(t & 15);
    int s = (int)src[e];
    int d = (int)dst[e];
    atomicAdd(&agg2[(long long)d * HID + c], h[(long long)s * HID + c]);
}

// per-graph node counts
__global__ void sage_gcount(const long long* __restrict__ batch,
                            float* __restrict__ gcnt) {
    int i = blockIdx.x * blockDim.x + threadIdx.x;
    if (i >= N_NODES) return;
    atomicAdd(&gcnt[(int)batch[i]], 1.0f);
}

// ---------------------------------------------------------------------------
// Layer 2 + fused graph pooling:
//   v = mean(agg2) @ w2_l^T + b2 + h @ w2_r^T   -> gsum[batch[node]] += v
// One wave per 16-node tile; 4 N-tiles x (4+4) K-steps of WMMA = 32 wmma/wave.
// ---------------------------------------------------------------------------
__global__ void sage_layer2(const float* __restrict__ hbuf,
                            const float* __restrict__ agg2,
                            const float* __restrict__ deg,
                            const long long* __restrict__ batch,
                            const float* __restrict__ w2l,   // [64][16]
                            const float* __restrict__ b2,    // [64]
                            const float* __restrict__ w2r,   // [64][16]
                            float* __restrict__ gsum) {
    int wave = (blockIdx.x * blockDim.x + threadIdx.x) >> 5;
    int lane = threadIdx.x & 31;
    if (wave >= N_TILES) return;
    int m     = lane & 15;
    int khalf = lane >> 4;
    int node_m = wave * 16 + m;
    float rdeg = 1.0f / fmaxf(deg[node_m], 1.0f);

    const float* aggrow = agg2 + (long long)node_m * HID;
    const float* hrow   = hbuf + (long long)node_m * HID;

    v8f acc[4] = {{}, {}, {}, {}};
#pragma unroll
    for (int ct = 0; ct < 4; ++ct) {
#pragma unroll
        for (int k = 0; k < HID; k += 4) {
            int ko = k + 2 * khalf;
            v2f a = *(const v2f*)(aggrow + ko);
            a.x *= rdeg; a.y *= rdeg;
            v2f b = *(const v2f*)(w2l + (ct * 16 + m) * HID + ko);
            acc[ct] = __builtin_amdgcn_wmma_f32_16x16x4_f32(false, a, false, b,
                                                            (short)0, acc[ct],
                                                            false, false);
        }
#pragma unroll
        for (int k = 0; k < HID; k += 4) {
            int ko = k + 2 * khalf;
            v2f a = *(const v2f*)(hrow + ko);
            v2f b = *(const v2f*)(w2r + (ct * 16 + m) * HID + ko);
            acc[ct] = __builtin_amdgcn_wmma_f32_16x16x4_f32(false, a, false, b,
                                                            (short)0, acc[ct],
                                                            false, false);
        }
    }

#pragma unroll
    for (int ct = 0; ct < 4; ++ct) {
        float bias = b2[ct * 16 + m];
#pragma unroll
        for (int j = 0; j < 8; ++j) {
            int row  = j + 8 * khalf;
            int node = wave * 16 + row;
            int g    = (int)batch[node];
            atomicAdd(&gsum[g * OUT_CH + ct * 16 + m], acc[ct][j] + bias);
        }
    }
}

__global__ void sage_finalize(const float* __restrict__ gsum,
                              const float* __restrict__ gcnt,
                              float* __restrict__ out) {
    int t = blockIdx.x * blockDim.x + threadIdx.x;
    if (t >= NUM_GRAPHS * OUT_CH) return;
    int g = t / OUT_CH;
    out[t] = gsum[t] / fmaxf(gcnt[g], 1.0f);
}

// ---------------------------------------------------------------------------
extern "C" void kernel_launch(void* const* d_in, const int* in_sizes, int n_in,
                              void* d_out, int out_size, void* d_ws, size_t ws_size,
                              hipStream_t stream) {
    const float*     x    = (const float*)d_in[0];
    const long long* ei   = (const long long*)d_in[1];     // int64 [2, E]
    const long long* batch= (const long long*)d_in[2];     // int64 [N]
    const float*     w1l  = (const float*)d_in[3];
    const float*     b1   = (const float*)d_in[4];
    const float*     w1r  = (const float*)d_in[5];
    const float*     w2l  = (const float*)d_in[6];
    const float*     b2   = (const float*)d_in[7];
    const float*     w2r  = (const float*)d_in[8];
    float* out = (float*)d_out;
    float* ws  = (float*)d_ws;

    const long long* src = ei;
    const long long* dst = ei + N_EDGES;

    // zero accumulation buffers (graph-capture-safe memset node)
    hipMemsetAsync(d_ws, 0, (size_t)WS_TOTAL * sizeof(float), stream);

    {   // scatter-add layer 1 (+ degree)
        long long nthr = (long long)N_EDGES * IN_CH;
        int blocks = (int)((nthr + 255) / 256);
        sage_scatter1<<<blocks, 256, 0, stream>>>(x, src, dst,
                                                  ws + WS_AGG1, ws + WS_DEG);
    }
    {   // layer 1 WMMA
        int waves  = N_TILES;
        int blocks = (waves * 32 + 255) / 256;
        sage_layer1<<<blocks, 256, 0, stream>>>(x, ws + WS_AGG1, ws + WS_DEG,
                                                w1l, b1, w1r, ws + WS_H);
    }
    {   // scatter-add layer 2
        long long nthr = (long long)N_EDGES * HID;
        int blocks = (int)((nthr + 255) / 256);
        sage_scatter2<<<blocks, 256, 0, stream>>>(ws + WS_H, src, dst,
                                                  ws + WS_AGG2);
    }
    {   // per-graph counts
        int blocks = (N_NODES + 255) / 256;
        sage_gcount<<<blocks, 256, 0, stream>>>(batch, ws + WS_GCNT);
    }
    {   // layer 2 WMMA + fused graph pooling
        int waves  = N_TILES;
        int blocks = (waves * 32 + 255) / 256;
        sage_layer2<<<blocks, 256, 0, stream>>>(ws + WS_H, ws + WS_AGG2,
                                                ws + WS_DEG, batch,
                                                w2l, b2, w2r, ws + WS_GSUM);
    }
    {   // divide by graph counts
        int blocks = (NUM_GRAPHS * OUT_CH + 255) / 256;
        sage_finalize<<<blocks, 256, 0, stream>>>(ws + WS_GSUM, ws + WS_GCNT, out);
    }
}